// Gene_ontology_network_75007308858110
// MI455X (gfx1250) — compile-verified
//
#include <hip/hip_runtime.h>

typedef float v2f __attribute__((ext_vector_type(2)));
typedef float v8f __attribute__((ext_vector_type(8)));

#define EPSN 1e-5f
#define CDIV(a,b) (((a)+(b)-1)/(b))

__device__ __forceinline__ int lb_i(const int* __restrict__ a, int n, int key) {
  int lo = 0, hi = n;
  while (lo < hi) { int mid = (lo + hi) >> 1; if (a[mid] < key) lo = mid + 1; else hi = mid; }
  return lo;
}

__device__ __forceinline__ float2 ld2(const float* p) {
  return *reinterpret_cast<const float2*>(p);
}

// ============ Generic WMMA f32 GEMM: C[R x O] = A[R x K] @ W[O x K]^T ============
// One wave per 16x16 tile. Unconditional clamped b64 loads + mask-multiply
// (no EXEC-branching around loads; EXEC stays all-ones for WMMA).
__global__ void gemm_nt_wmma(const float* __restrict__ A, const float* __restrict__ W,
                             float* __restrict__ C, int R, int K, int O) {
  int tilesO = (O + 15) >> 4;
  int t  = blockIdx.x;
  int tr = t / tilesO, tc = t - tr * tilesO;
  int lane = threadIdx.x;                 // 0..31 (wave32)
  int row0 = tr << 4, col0 = tc << 4;
  int m    = lane & 15;
  int ksel = (lane >> 4) << 1;            // 0 for lanes 0-15, 2 for lanes 16-31
  int rA = row0 + m;
  int oB = col0 + m;
  bool rOk = rA < R;
  bool oOk = oB < O;
  float rm = rOk ? 1.f : 0.f;
  float om = oOk ? 1.f : 0.f;
  const float* Ap = A + (size_t)(rOk ? rA : 0) * K + ksel;
  const float* Wp = W + (size_t)(oOk ? oB : 0) * K + ksel;
  v8f acc = {0.f,0.f,0.f,0.f,0.f,0.f,0.f,0.f};
  for (int k0 = 0; k0 < K; k0 += 4) {
    float2 a2 = ld2(Ap + k0);
    float2 w2 = ld2(Wp + k0);
    v2f av, bv;
    av.x = a2.x * rm; av.y = a2.y * rm;
    bv.x = w2.x * om; bv.y = w2.y * om;
    acc = __builtin_amdgcn_wmma_f32_16x16x4_f32(false, av, false, bv,
                                                (short)0, acc, false, false);
  }
  int colw  = col0 + m;
  int rbase = row0 + ((lane >> 4) << 3);
  if (colw < O) {
#pragma unroll
    for (int r = 0; r < 8; ++r) {
      int row = rbase + r;
      if (row < R) C[(size_t)row * O + colw] = acc[r];
    }
  }
}

// ============ Specialized WMMA GEMM for O == 64 ============
// One wave computes a full 16x64 output strip: A fragment reused across 4 WMMAs.
__global__ void gemm_nt_wmma_o64(const float* __restrict__ A, const float* __restrict__ W,
                                 float* __restrict__ C, int R, int K) {
  int tr = blockIdx.x;
  int lane = threadIdx.x;
  int m    = lane & 15;
  int ksel = (lane >> 4) << 1;
  int rA = (tr << 4) + m;
  bool rOk = rA < R;
  float rm = rOk ? 1.f : 0.f;
  const float* Ap = A + (size_t)(rOk ? rA : 0) * K + ksel;
  const float* Wp = W + (size_t)m * K + ksel;   // col tile c adds 16*K
  size_t wstride = (size_t)16 * K;
  v8f a0 = {0.f,0.f,0.f,0.f,0.f,0.f,0.f,0.f};
  v8f a1 = a0, a2 = a0, a3 = a0;
  for (int k0 = 0; k0 < K; k0 += 4) {
    float2 af = ld2(Ap + k0);
    v2f av; av.x = af.x * rm; av.y = af.y * rm;
    float2 w0 = ld2(Wp + k0);
    float2 w1 = ld2(Wp + wstride + k0);
    float2 w2 = ld2(Wp + 2 * wstride + k0);
    float2 w3 = ld2(Wp + 3 * wstride + k0);
    v2f b0; b0.x = w0.x; b0.y = w0.y;
    v2f b1; b1.x = w1.x; b1.y = w1.y;
    v2f b2; b2.x = w2.x; b2.y = w2.y;
    v2f b3; b3.x = w3.x; b3.y = w3.y;
    a0 = __builtin_amdgcn_wmma_f32_16x16x4_f32(false, av, false, b0, (short)0, a0, false, false);
    a1 = __builtin_amdgcn_wmma_f32_16x16x4_f32(false, av, false, b1, (short)0, a1, false, false);
    a2 = __builtin_amdgcn_wmma_f32_16x16x4_f32(false, av, false, b2, (short)0, a2, false, false);
    a3 = __builtin_amdgcn_wmma_f32_16x16x4_f32(false, av, false, b3, (short)0, a3, false, false);
  }
  int rbase = (tr << 4) + ((lane >> 4) << 3);
#pragma unroll
  for (int r = 0; r < 8; ++r) {
    int row = rbase + r;
    if (row < R) {
      float* Crow = C + (size_t)row * 64 + m;
      Crow[0]  = a0[r];
      Crow[16] = a1[r];
      Crow[32] = a2[r];
      Crow[48] = a3[r];
    }
  }
}

// ============ x0[b,go,f] = sum_e{Ag_row==go} data[b, Ag_col[e]] * t[f,e] ============
__global__ void gene_gather(const float* __restrict__ data, const float* __restrict__ t,
                            const int* __restrict__ row, const int* __restrict__ col,
                            int nnz, float* __restrict__ x0) {
  int go = blockIdx.x;                    // 0..11999
  int lo = lb_i(row, nnz, go);
  int hi = lb_i(row, nnz, go + 1);
  int tid = threadIdx.x;                  // 64 threads: b = tid>>3, f = tid&7
  int b = tid >> 3, f = tid & 7;
  float acc = 0.f;
  const float* db = data + (size_t)b * 20000;
  const float* tf = t + (size_t)f * nnz;
  for (int e = lo; e < hi; ++e) acc += db[col[e]] * tf[e];
  x0[((size_t)b * 12000 + go) * 8 + f] = acc;
}

// ============ per-(b,node) attention scalars ============
__global__ void enc_att_scalars(const float* __restrict__ x_in, const float* __restrict__ x_s,
                                const float* __restrict__ w_att_in, const float* __restrict__ w_att_s,
                                float* __restrict__ a_src, float* __restrict__ a_dst,
                                float* __restrict__ gate, int total) {
  int i = blockIdx.x * blockDim.x + threadIdx.x;
  if (i >= total) return;
  const float* xi = x_in + (size_t)i * 64;
  const float* xs = x_s  + (size_t)i * 64;
  float as = 0.f, ad = 0.f, gs = 0.f;
#pragma unroll 4
  for (int o = 0; o < 64; ++o) {
    float v = xi[o];
    as += v * w_att_in[o];
    ad += v * w_att_in[64 + o];
    gs += xs[o] * w_att_s[o];
  }
  a_src[i] = as; a_dst[i] = ad;
  gate[i] = 1.f / (1.f + expf(-gs));
}

// ============ encoder softmax aggregation (src sorted) ============
__global__ void enc_aggregate(const float* __restrict__ x_in,
                              const float* __restrict__ a_src, const float* __restrict__ a_dst,
                              const int* __restrict__ src, const int* __restrict__ dst,
                              int E, int n, float* __restrict__ agg) {
  int s  = blockIdx.x;
  int lo = lb_i(src, E, s);
  int hi = lb_i(src, E, s + 1);
  int t = threadIdx.x;                    // 512 threads: b = t>>6, f = t&63
  int b = t >> 6, f = t & 63;
  __shared__ float denominv[8];
  __shared__ float vbuf[64 * 8];
  __shared__ int   dstbuf[64];
  if (t < 8) {
    float sum = 0.f;
    float as = a_src[(size_t)t * n + s];
    for (int e = lo; e < hi; ++e)
      sum += expf(tanhf(as + a_dst[(size_t)t * n + dst[e]]));
    denominv[t] = (hi > lo) ? 1.f / sum : 0.f;
  }
  __syncthreads();
  float acc = 0.f;
  float asb = a_src[(size_t)b * n + s];
  for (int done = 0; done < hi - lo; done += 64) {
    int cnt = hi - lo - done; if (cnt > 64) cnt = 64;
    for (int i = t; i < cnt * 8; i += 512) {
      int ei = i >> 3, b2 = i & 7;
      int e  = lo + done + ei;
      float as2 = (b2 == b) ? asb : a_src[(size_t)b2 * n + s];
      float v = expf(tanhf(as2 + a_dst[(size_t)b2 * n + dst[e]]));
      vbuf[ei * 8 + b2] = v * denominv[b2];
    }
    if (t < cnt) dstbuf[t] = dst[lo + done + t];
    __syncthreads();
    for (int j = 0; j < cnt; ++j)
      acc += vbuf[j * 8 + b] * x_in[((size_t)b * n + dstbuf[j]) * 64 + f];
    __syncthreads();
  }
  agg[((size_t)b * n + s) * 64 + f] = acc;
}

// ============ encoder combine: pre = agg + x_s * gate ============
__global__ void enc_combine(const float* __restrict__ agg, const float* __restrict__ x_s,
                            const float* __restrict__ gate, float* __restrict__ pre, size_t total) {
  size_t i = (size_t)blockIdx.x * blockDim.x + threadIdx.x;
  if (i >= total) return;
  pre[i] = agg[i] + x_s[i] * gate[i >> 6];   // F = 64
}

// ============ LayerNorm over node axis + ReLU ============
__global__ void ln_nodes_relu(const float* __restrict__ pre, const float* __restrict__ g,
                              const float* __restrict__ beta, float* __restrict__ out,
                              int n, int F) {
  int blk = blockIdx.x;
  int b = blk / F, f = blk - b * F;
  const float* base = pre + (size_t)b * n * F + f;
  float*       obase = out + (size_t)b * n * F + f;
  __shared__ float s1[256], s2[256], stats[2];
  float sum = 0.f, sq = 0.f;
  for (int i = threadIdx.x; i < n; i += 256) {
    float v = base[(size_t)i * F];
    sum += v; sq += v * v;
  }
  s1[threadIdx.x] = sum; s2[threadIdx.x] = sq;
  __syncthreads();
  for (int off = 128; off > 0; off >>= 1) {
    if (threadIdx.x < off) { s1[threadIdx.x] += s1[threadIdx.x + off]; s2[threadIdx.x] += s2[threadIdx.x + off]; }
    __syncthreads();
  }
  if (threadIdx.x == 0) {
    float mu  = s1[0] / n;
    float var = s2[0] / n - mu * mu;
    stats[0] = mu; stats[1] = rsqrtf(var + EPSN);
  }
  __syncthreads();
  float mu = stats[0], rs = stats[1];
  for (int i = threadIdx.x; i < n; i += 256) {
    float v = (base[(size_t)i * F] - mu) * rs * g[i] + beta[i];
    obase[(size_t)i * F] = fmaxf(v, 0.f);
  }
}

// ============ node slice copy ============
__global__ void slice_nodes(const float* __restrict__ src, float* __restrict__ dst,
                            int n_src, int off, int n_dst, int F, size_t total) {
  size_t idx = (size_t)blockIdx.x * blockDim.x + threadIdx.x;
  if (idx >= total) return;
  int f  = (int)(idx % F);
  size_t r = idx / F;
  int nd = (int)(r % n_dst);
  int b  = (int)(r / n_dst);
  dst[idx] = src[((size_t)b * n_src + off + nd) * F + f];
}

// ============ BatchNorm 2D over batch axis + ReLU ============
__global__ void bn2d_relu(const float* __restrict__ x, const float* __restrict__ g,
                          const float* __restrict__ beta, float* __restrict__ out,
                          int Bsz, int n) {
  int c = blockIdx.x * blockDim.x + threadIdx.x;
  if (c >= n) return;
  float sum = 0.f, sq = 0.f;
  for (int b = 0; b < Bsz; ++b) { float v = x[(size_t)b * n + c]; sum += v; sq += v * v; }
  float mu = sum / Bsz, var = sq / Bsz - mu * mu;
  float rs = rsqrtf(var + EPSN);
  float gg = g[c], bb = beta[c];
  for (int b = 0; b < Bsz; ++b) {
    float v = (x[(size_t)b * n + c] - mu) * rs * gg + bb;
    out[(size_t)b * n + c] = fmaxf(v, 0.f);
  }
}

// ============ BatchNorm 3D (stats over batch & channel) + ReLU ============
__global__ void bn3d_relu(const float* __restrict__ x, const float* __restrict__ g,
                          const float* __restrict__ beta, float* __restrict__ out,
                          int n, int D) {
  int c = blockIdx.x * blockDim.x + threadIdx.x;
  if (c >= n) return;
  float sum = 0.f, sq = 0.f;
  for (int b = 0; b < 8; ++b)
    for (int d = 0; d < 5; ++d) {
      float v = x[((size_t)b * n + c) * D + d];
      sum += v; sq += v * v;
    }
  float mu = sum / 40.f, var = sq / 40.f - mu * mu;
  float rs = rsqrtf(var + EPSN);
  float gg = g[c], bb = beta[c];
  for (int b = 0; b < 8; ++b)
    for (int d = 0; d < 5; ++d) {
      float v = (x[((size_t)b * n + c) * D + d] - mu) * rs * gg + bb;
      out[((size_t)b * n + c) * D + d] = fmaxf(v, 0.f);
    }
}

__global__ void fill_zero(float* __restrict__ p, size_t n) {
  size_t i = (size_t)blockIdx.x * blockDim.x + threadIdx.x;
  if (i < n) p[i] = 0.f;
}

__global__ void deg_count(const int* __restrict__ src, int E, float* __restrict__ deg) {
  int e = blockIdx.x * blockDim.x + threadIdx.x;
  if (e < E) atomicAdd(&deg[src[e]], 1.f);
}

// ============ decoder scatter: agg[b,src,:] += x_out[b,dst,:] / deg[src] ============
__global__ void dec_scatter(const float* __restrict__ x_out,
                            const int* __restrict__ src, const int* __restrict__ dst,
                            const float* __restrict__ deg,
                            int E, int n_in, int n_rows, int F, float* __restrict__ agg) {
  int idx = blockIdx.x * blockDim.x + threadIdx.x;
  if (idx >= E * F) return;
  int e = idx / F, f = idx - e * F;
  int s = src[e], d = dst[e];
  float inv = 1.f / deg[s];
  for (int b = 0; b < 8; ++b)
    atomicAdd(&agg[((size_t)b * n_rows + s) * F + f],
              x_out[((size_t)b * n_in + d) * F + f] * inv);
}

// ============ decoder combine: pre = agg; pre[:, off:, :] += x_s ============
__global__ void dec_combine(const float* __restrict__ agg, const float* __restrict__ x_s,
                            float* __restrict__ pre, int n_rows, int off, int F, size_t total) {
  size_t idx = (size_t)blockIdx.x * blockDim.x + threadIdx.x;
  if (idx >= total) return;
  int f = (int)(idx % F);
  size_t r = idx / F;
  int s = (int)(r % n_rows);
  int b = (int)(r / n_rows);
  float v = agg[idx];
  if (s >= off) v += x_s[((size_t)b * (n_rows - off) + (s - off)) * F + f];
  pre[idx] = v;
}

// ============ xD[b, Ag_col[e]] += out_D[b, Ag_row[e]] * t_D[e] ============
__global__ void gene_scatter(const float* __restrict__ out_D, const float* __restrict__ t_D,
                             const int* __restrict__ row, const int* __restrict__ col,
                             int nnz, float* __restrict__ xD) {
  int e = blockIdx.x * blockDim.x + threadIdx.x;
  if (e >= nnz) return;
  int r = row[e], c = col[e];
  float td = t_D[e];
  for (int b = 0; b < 8; ++b)
    atomicAdd(&xD[(size_t)b * 20000 + c], out_D[(size_t)b * 12000 + r] * td);
}

extern "C" void kernel_launch(void* const* d_in, const int* in_sizes, int n_in,
                              void* d_out, int out_size, void* d_ws, size_t ws_size,
                              hipStream_t stream) {
  const float* data      = (const float*)d_in[0];
  const float* t         = (const float*)d_in[1];
  const float* t_D       = (const float*)d_in[2];
  const float* w_inc0    = (const float*)d_in[3];
  const float* w_inc1    = (const float*)d_in[4];
  const float* w_s0      = (const float*)d_in[5];
  const float* w_s1      = (const float*)d_in[6];
  const float* w_att_in0 = (const float*)d_in[7];
  const float* w_att_in1 = (const float*)d_in[8];
  const float* w_att_s0  = (const float*)d_in[9];
  const float* w_att_s1  = (const float*)d_in[10];
  const float* gb_g0 = (const float*)d_in[11];
  const float* gb_b0 = (const float*)d_in[12];
  const float* gb_g1 = (const float*)d_in[13];
  const float* gb_b1 = (const float*)d_in[14];
  const float* w_out0  = (const float*)d_in[15];
  const float* w_out1  = (const float*)d_in[16];
  const float* w_sout0 = (const float*)d_in[17];
  const float* w_sout1 = (const float*)d_in[18];
  const float* gbd_g0 = (const float*)d_in[19];
  const float* gbd_b0 = (const float*)d_in[20];
  const float* gbd_g1 = (const float*)d_in[21];
  const float* gbd_b1 = (const float*)d_in[22];
  const float* catt_w    = (const float*)d_in[23];
  const float* catt_bn_g = (const float*)d_in[24];
  const float* catt_bn_b = (const float*)d_in[25];
  const float* conc_w  = (const float*)d_in[26];
  const float* b_bn_g  = (const float*)d_in[27];
  const float* b_bn_b  = (const float*)d_in[28];
  const float* concd_w = (const float*)d_in[29];
  const float* bd_bn_g = (const float*)d_in[30];
  const float* bd_bn_b = (const float*)d_in[31];
  const float* lat_w1    = (const float*)d_in[32];
  const float* lat_bn1_g = (const float*)d_in[33];
  const float* lat_bn1_b = (const float*)d_in[34];
  const float* lat_w2    = (const float*)d_in[35];
  const float* lat_bn2_g = (const float*)d_in[36];
  const float* lat_bn2_b = (const float*)d_in[37];
  const int* Ag_row   = (const int*)d_in[38];
  const int* Ag_col   = (const int*)d_in[39];
  const int* enc_src0 = (const int*)d_in[40];
  const int* enc_dst0 = (const int*)d_in[41];
  const int* enc_src1 = (const int*)d_in[42];
  const int* enc_dst1 = (const int*)d_in[43];
  const int* dec_src0 = (const int*)d_in[44];
  const int* dec_dst0 = (const int*)d_in[45];
  const int* dec_src1 = (const int*)d_in[46];
  const int* dec_dst1 = (const int*)d_in[47];

  const int NNZ = in_sizes[2];
  const int E0  = in_sizes[40], E1 = in_sizes[42];
  const int Ed0 = in_sizes[44], Ed1 = in_sizes[46];
  float* out = (float*)d_out;
  float* latent_out = out;            // 128
  float* xD_out     = out + 128;      // 160000
  float* atten_out  = out + 160128;   // 160000

  // ---- workspace arena (floats) ----
  const size_t NF = 6144000;          // 8*12000*64
  float* ws   = (float*)d_ws;
  float* bufA = ws;                   // x_in / x_out
  float* bufB = bufA + NF;            // x_s
  float* bufC = bufB + NF;            // aggregation
  float* bufD = bufC + NF;            // pre-LN
  float* bufE = bufD + NF;            // stage outputs
  float* bufX = bufE + NF;            // stage inputs (<= 4.1M floats)
  float* sm     = bufX + 4100000;
  float* aSrc   = sm;                 // 96000
  float* aDst   = aSrc + 96000;       // 96000
  float* gateB  = aDst + 96000;       // 96000
  float* degB   = gateB + 96000;      // 12000
  float* inpOut = degB + 12000;       // 32000
  float* outD   = inpOut + 32000;     // 96000
  float* tmpS   = outD + 96000;       // 160000
  float* hpre   = tmpS + 160000;      // 256
  float* hrel   = hpre + 256;         // 256
  float* lpre   = hrel + 256;         // 128

  auto gemm = [&](const float* A, const float* W, float* C, int R, int K, int O) {
    if (O == 64) {
      gemm_nt_wmma_o64<<<CDIV(R, 16), 32, 0, stream>>>(A, W, C, R, K);
    } else {
      int blocks = CDIV(R, 16) * CDIV(O, 16);
      gemm_nt_wmma<<<blocks, 32, 0, stream>>>(A, W, C, R, K, O);
    }
  };

  // 1) x0 = segment_sum(data gather * t) : (8,12000,8) in bufX
  gene_gather<<<12000, 64, 0, stream>>>(data, t, Ag_row, Ag_col, NNZ, bufX);

  // 2) encoder level 0 (n=12000, K=8)
  gemm(bufX, w_inc0, bufA, 96000, 8, 64);
  gemm(bufX, w_s0,   bufB, 96000, 8, 64);
  enc_att_scalars<<<CDIV(96000,256),256,0,stream>>>(bufA, bufB, w_att_in0, w_att_s0, aSrc, aDst, gateB, 96000);
  enc_aggregate<<<12000, 512, 0, stream>>>(bufA, aSrc, aDst, enc_src0, enc_dst0, E0, 12000, bufC);
  enc_combine<<<CDIV(6144000,256),256,0,stream>>>(bufC, bufB, gateB, bufD, (size_t)6144000);
  ln_nodes_relu<<<8*64, 256, 0, stream>>>(bufD, gb_g0, gb_b0, bufE, 12000, 64);

  // 3) slice [:,4000:,:] -> bufX (8,8000,64)
  slice_nodes<<<CDIV(4096000,256),256,0,stream>>>(bufE, bufX, 12000, 4000, 8000, 64, (size_t)4096000);

  // 4) encoder level 1 (n=8000, K=64)
  gemm(bufX, w_inc1, bufA, 64000, 64, 64);
  gemm(bufX, w_s1,   bufB, 64000, 64, 64);
  enc_att_scalars<<<CDIV(64000,256),256,0,stream>>>(bufA, bufB, w_att_in1, w_att_s1, aSrc, aDst, gateB, 64000);
  enc_aggregate<<<8000, 512, 0, stream>>>(bufA, aSrc, aDst, enc_src1, enc_dst1, E1, 8000, bufC);
  enc_combine<<<CDIV(4096000,256),256,0,stream>>>(bufC, bufB, gateB, bufD, (size_t)4096000);
  ln_nodes_relu<<<8*64, 256, 0, stream>>>(bufD, gb_g1, gb_b1, bufE, 8000, 64);

  // 5) slice [:,4000:,:] -> bufX (8,4000,64)
  slice_nodes<<<CDIV(2048000,256),256,0,stream>>>(bufE, bufX, 8000, 4000, 4000, 64, (size_t)2048000);

  // 6) atten head: (32000x64)@(64x5) -> bn3d -> d_out
  gemm(bufX, catt_w, tmpS, 32000, 64, 5);
  bn3d_relu<<<CDIV(4000,64),64,0,stream>>>(tmpS, catt_bn_g, catt_bn_b, atten_out, 4000, 5);

  // 7) conc head -> inp_out (8,4000)
  gemm(bufX, conc_w, tmpS, 32000, 64, 1);
  bn2d_relu<<<CDIV(4000,128),128,0,stream>>>(tmpS, b_bn_g, b_bn_b, inpOut, 8, 4000);

  // 8) decoder level 0: n_in=4000 -> n_rows=8000, offset=4000
  gemm(bufX, w_out0,  bufA, 32000, 64, 64);
  gemm(bufX, w_sout0, bufB, 32000, 64, 64);
  fill_zero<<<CDIV(8000,256),256,0,stream>>>(degB, (size_t)8000);
  deg_count<<<CDIV(Ed0,256),256,0,stream>>>(dec_src0, Ed0, degB);
  fill_zero<<<CDIV(4096000,256),256,0,stream>>>(bufC, (size_t)4096000);
  dec_scatter<<<CDIV(Ed0*64,256),256,0,stream>>>(bufA, dec_src0, dec_dst0, degB, Ed0, 4000, 8000, 64, bufC);
  dec_combine<<<CDIV(4096000,256),256,0,stream>>>(bufC, bufB, bufD, 8000, 4000, 64, (size_t)4096000);
  ln_nodes_relu<<<8*64, 256, 0, stream>>>(bufD, gbd_g0, gbd_b0, bufE, 8000, 64);

  // 9) decoder level 1: n_in=8000 -> n_rows=12000, offset=4000, F_out=8
  gemm(bufE, w_out1,  bufA, 64000, 64, 8);
  gemm(bufE, w_sout1, bufB, 64000, 64, 8);
  fill_zero<<<CDIV(12000,256),256,0,stream>>>(degB, (size_t)12000);
  deg_count<<<CDIV(Ed1,256),256,0,stream>>>(dec_src1, Ed1, degB);
  fill_zero<<<CDIV(768000,256),256,0,stream>>>(bufC, (size_t)768000);
  dec_scatter<<<CDIV(Ed1*8,256),256,0,stream>>>(bufA, dec_src1, dec_dst1, degB, Ed1, 8000, 12000, 8, bufC);
  dec_combine<<<CDIV(768000,256),256,0,stream>>>(bufC, bufB, bufD, 12000, 4000, 8, (size_t)768000);
  ln_nodes_relu<<<8*8, 256, 0, stream>>>(bufD, gbd_g1, gbd_b1, bufX, 12000, 8);

  // 10) concd head -> out_D (8,12000)
  gemm(bufX, concd_w, tmpS, 96000, 8, 1);
  bn2d_relu<<<CDIV(12000,128),128,0,stream>>>(tmpS, bd_bn_g, bd_bn_b, outD, 8, 12000);

  // 11) x_D scatter to genes
  fill_zero<<<CDIV(160000,256),256,0,stream>>>(xD_out, (size_t)160000);
  gene_scatter<<<CDIV(NNZ,256),256,0,stream>>>(outD, t_D, Ag_row, Ag_col, NNZ, xD_out);

  // 12) latent MLP: (8x4000)@(4000x32) -> bn -> (8x32)@(32x16) -> bn -> latent
  gemm(inpOut, lat_w1, hpre, 8, 4000, 32);
  bn2d_relu<<<1, 64, 0, stream>>>(hpre, lat_bn1_g, lat_bn1_b, hrel, 8, 32);
  gemm(hrel, lat_w2, lpre, 8, 32, 16);
  bn2d_relu<<<1, 32, 0, stream>>>(lpre, lat_bn2_g, lat_bn2_b, latent_out, 8, 16);
}